// GCNNet_60309930770452
// MI455X (gfx1250) — compile-verified
//
#include <hip/hip_runtime.h>

#define HIDDEN 128
#define NUM_LAYERS 6

typedef __attribute__((ext_vector_type(2))) float v2f;
typedef __attribute__((ext_vector_type(8))) float v8f;

// ---------------- CSR construction ----------------

__global__ void k_init_deg(int* __restrict__ deg, int n) {
  int i = blockIdx.x * blockDim.x + threadIdx.x;
  if (i < n) deg[i] = 1;  // self-loop contributes 1
}

__global__ void k_count_dst(const int* __restrict__ dst, int* __restrict__ deg, int e) {
  int i = blockIdx.x * blockDim.x + threadIdx.x;
  if (i < e) atomicAdd(&deg[dst[i]], 1);
}

__global__ void k_dinv(const int* __restrict__ deg, float* __restrict__ dinv, int n) {
  int i = blockIdx.x * blockDim.x + threadIdx.x;
  if (i < n) dinv[i] = rsqrtf((float)deg[i]);  // deg >= 1 always (self-loop)
}

// single-block Hillis-Steele scan over chunks of 1024 (100k elems, negligible cost)
__global__ void k_scan(const int* __restrict__ deg, int* __restrict__ rowptr, int n) {
  __shared__ int s[1024];
  __shared__ int carry;
  int tid = threadIdx.x;
  if (tid == 0) { carry = 0; rowptr[0] = 0; }
  __syncthreads();
  for (int base = 0; base < n; base += 1024) {
    int i = base + tid;
    int v = (i < n) ? deg[i] : 0;
    s[tid] = v;
    __syncthreads();
    for (int off = 1; off < 1024; off <<= 1) {
      int t = (tid >= off) ? s[tid - off] : 0;
      __syncthreads();
      s[tid] += t;
      __syncthreads();
    }
    int incl = s[tid] + carry;
    if (i < n) rowptr[i + 1] = incl;
    __syncthreads();
    if (tid == 1023) carry = incl;
    __syncthreads();
  }
}

__global__ void k_fill_self(const float* __restrict__ dinv, const int* __restrict__ rowptr,
                            int* __restrict__ col, float* __restrict__ wgt,
                            int* __restrict__ cursor, int n) {
  int i = blockIdx.x * blockDim.x + threadIdx.x;
  if (i < n) {
    int p = rowptr[i];
    col[p] = i;
    float d = dinv[i];
    wgt[p] = d * d;
    cursor[i] = 1;  // slot 0 taken by self-loop
  }
}

__global__ void k_fill_edges(const int* __restrict__ src, const int* __restrict__ dst,
                             const float* __restrict__ dinv, const int* __restrict__ rowptr,
                             int* __restrict__ cursor, int* __restrict__ col,
                             float* __restrict__ wgt, int e) {
  int i = blockIdx.x * blockDim.x + threadIdx.x;
  if (i < e) {
    int d = dst[i], s = src[i];
    int off = atomicAdd(&cursor[d], 1);
    int p = rowptr[d] + off;
    col[p] = s;
    wgt[p] = dinv[s] * dinv[d];
  }
}

// ---------------- embedding lookup ----------------

__global__ void k_embed(const int* __restrict__ z, const float* __restrict__ emb,
                        float* __restrict__ x, int n) {
  int i = blockIdx.x * blockDim.x + threadIdx.x;
  if (i < n * HIDDEN) {
    int node = i >> 7, c = i & 127;
    x[i] = emb[z[node] * HIDDEN + c];
  }
}

// ---------------- fp32 WMMA GEMM: H[N,128] = X[N,128] @ W[128,128] ----------------
// One wave produces a 16x128 output stripe (8 column tiles of 16x16),
// accumulating K=128 in 32 steps of V_WMMA_F32_16X16X4_F32.
// A layout (16x4 f32): lane 0-15 row M=lane, v[0]=K0 v[1]=K1; lanes 16-31: K2/K3.
// B layout (4x16 f32): lane 0-15 col N=lane, v[0]=K0 v[1]=K1; lanes 16-31: K2/K3.
// W is staged in LDS K-pair-major: pair (W[2k2][c], W[2k2+1][c]) contiguous, so
// every B operand is ONE aligned ds_load_b64 straight into the WMMA source pair.
// Each k2 row is skewed by 128B (stride 288 floats, 1152B; 1152 % 256 == 128) so
// lanes 0-15 hit banks 0-31 and lanes 16-31 hit banks 32-63: conflict-free b64.

#define WROW 288  // floats per k2 row: 128 cols * 2 + 32-float skew

__global__ void __launch_bounds__(256) k_gemm(const float* __restrict__ X,
                                              const float* __restrict__ W,
                                              float* __restrict__ H, int n) {
  __shared__ float sW[64 * WROW];  // 72 KB of the 320 KB WGP LDS
  for (int i = threadIdx.x; i < HIDDEN * HIDDEN; i += 256) {
    int k = i >> 7, c = i & 127;
    sW[(k >> 1) * WROW + c * 2 + (k & 1)] = W[i];
  }
  __syncthreads();

  int wave = threadIdx.x >> 5;
  int lane = threadIdx.x & 31;
  int half = lane >> 4;
  int lr   = lane & 15;
  int m0 = (blockIdx.x * 8 + wave) * 16;
  if (m0 >= n) return;  // wave-uniform: EXEC stays all-ones for WMMA

  const v8f zero = {0.f, 0.f, 0.f, 0.f, 0.f, 0.f, 0.f, 0.f};
  v8f acc[8];
#pragma unroll
  for (int t = 0; t < 8; ++t) acc[t] = zero;

  const float* xr = X + (size_t)(m0 + lr) * HIDDEN + 2 * half;
  for (int k0 = 0; k0 < HIDDEN; k0 += 4) {
    v2f a = *(const v2f*)(xr + k0);                 // global_load_b64, 8B aligned
    const float* bp = sW + ((k0 >> 1) + half) * WROW + lr * 2;
#pragma unroll
    for (int t = 0; t < 8; ++t) {
      v2f b = *(const v2f*)(bp + t * 32);           // ds_load_b64, imm offset t*128B
      acc[t] = __builtin_amdgcn_wmma_f32_16x16x4_f32(
          false, a, false, b, (short)0, acc[t], false, false);
    }
  }
#pragma unroll
  for (int t = 0; t < 8; ++t) {
#pragma unroll
    for (int v = 0; v < 8; ++v) {
      H[(size_t)(m0 + half * 8 + v) * HIDDEN + t * 16 + lr] = acc[t][v];
    }
  }
}

// ---------------- per-node gather-reduce (no float atomics) ----------------
// One wave per node; each lane owns a float4 slice of the 128-wide row, so
// every neighbor row is one contiguous 512B burst — served from 192MB L2.

__global__ void __launch_bounds__(256) k_aggregate(const float* __restrict__ H,
                                                   const int* __restrict__ rowptr,
                                                   const int* __restrict__ col,
                                                   const float* __restrict__ wgt,
                                                   const float* __restrict__ bias,
                                                   float* __restrict__ Xo, int n) {
  int node = blockIdx.x * (blockDim.x >> 5) + (threadIdx.x >> 5);
  if (node >= n) return;
  int lane = threadIdx.x & 31;
  float4 acc = make_float4(0.f, 0.f, 0.f, 0.f);
  int e0 = rowptr[node], e1 = rowptr[node + 1];
  for (int e = e0; e < e1; ++e) {
    int s = col[e];
    float w = wgt[e];
    float4 h = ((const float4*)(H + (size_t)s * HIDDEN))[lane];
    acc.x += w * h.x; acc.y += w * h.y; acc.z += w * h.z; acc.w += w * h.w;
  }
  float4 b = ((const float4*)bias)[lane];
  acc.x = fmaxf(acc.x + b.x, 0.f);
  acc.y = fmaxf(acc.y + b.y, 0.f);
  acc.z = fmaxf(acc.z + b.z, 0.f);
  acc.w = fmaxf(acc.w + b.w, 0.f);
  ((float4*)(Xo + (size_t)node * HIDDEN))[lane] = acc;
}

// ---------------- global_add_pool + head MLP ----------------

__global__ void k_zero(float* __restrict__ p, int total) {
  int i = blockIdx.x * blockDim.x + threadIdx.x;
  if (i < total) p[i] = 0.f;
}

__global__ void k_pool(const float* __restrict__ X, const int* __restrict__ batch,
                       float* __restrict__ G, int n) {
  int i = blockIdx.x * blockDim.x + threadIdx.x;
  if (i < n * HIDDEN) {
    int node = i >> 7, c = i & 127;
    atomicAdd(&G[batch[node] * HIDDEN + c], X[i]);
  }
}

__global__ void __launch_bounds__(128) k_mlp(const float* __restrict__ G,
                                             const float* __restrict__ W1,
                                             const float* __restrict__ b1,
                                             const float* __restrict__ W2,
                                             const float* __restrict__ b2,
                                             float* __restrict__ out) {
  __shared__ float sg[HIDDEN];
  __shared__ float sh[HIDDEN];
  int g = blockIdx.x;
  int t = threadIdx.x;
  sg[t] = G[(size_t)g * HIDDEN + t];
  __syncthreads();
  float acc = b1[t];
#pragma unroll 8
  for (int k = 0; k < HIDDEN; ++k) acc += sg[k] * W1[k * HIDDEN + t];
  acc = fmaxf(acc, 0.f);
  sh[t] = acc * W2[t];
  __syncthreads();
  for (int off = 64; off > 0; off >>= 1) {
    if (t < off) sh[t] += sh[t + off];
    __syncthreads();
  }
  if (t == 0) out[g] = sh[0] + b2[0];
}

// ---------------- launcher ----------------

extern "C" void kernel_launch(void* const* d_in, const int* in_sizes, int n_in,
                              void* d_out, int out_size, void* d_ws, size_t ws_size,
                              hipStream_t stream) {
  const int*   z     = (const int*)d_in[0];
  const int*   ei    = (const int*)d_in[1];  // [2, E] row-major
  const int*   batch = (const int*)d_in[2];
  const float* emb   = (const float*)d_in[3];
  const float* Ws    = (const float*)d_in[4];
  const float* bs    = (const float*)d_in[5];
  const float* W1    = (const float*)d_in[6];
  const float* b1    = (const float*)d_in[7];
  const float* W2    = (const float*)d_in[8];
  const float* b2    = (const float*)d_in[9];
  float* out = (float*)d_out;

  int n  = in_sizes[0];
  int e  = in_sizes[1] / 2;
  int ng = out_size;
  const int* src = ei;
  const int* dst = ei + e;

  char* ws = (char*)d_ws;
  size_t off = 0;
  auto alloc = [&](size_t bytes) -> char* {
    char* p = ws + off;
    off += (bytes + 255) & ~(size_t)255;
    return p;
  };
  float* X      = (float*)alloc((size_t)n * HIDDEN * sizeof(float));
  float* H      = (float*)alloc((size_t)n * HIDDEN * sizeof(float));
  float* G      = (float*)alloc((size_t)ng * HIDDEN * sizeof(float));
  int*   deg    = (int*)  alloc((size_t)n * sizeof(int));
  float* dinv   = (float*)alloc((size_t)n * sizeof(float));
  int*   rowptr = (int*)  alloc((size_t)(n + 1) * sizeof(int));
  int*   cursor = (int*)  alloc((size_t)n * sizeof(int));
  int    nnz    = e + n;
  int*   col    = (int*)  alloc((size_t)nnz * sizeof(int));
  float* wgt    = (float*)alloc((size_t)nnz * sizeof(float));
  (void)ws_size; (void)n_in;

  const int B = 256;
  k_init_deg<<<(n + B - 1) / B, B, 0, stream>>>(deg, n);
  k_count_dst<<<(e + B - 1) / B, B, 0, stream>>>(dst, deg, e);
  k_dinv<<<(n + B - 1) / B, B, 0, stream>>>(deg, dinv, n);
  k_scan<<<1, 1024, 0, stream>>>(deg, rowptr, n);
  k_fill_self<<<(n + B - 1) / B, B, 0, stream>>>(dinv, rowptr, col, wgt, cursor, n);
  k_fill_edges<<<(e + B - 1) / B, B, 0, stream>>>(src, dst, dinv, rowptr, cursor, col, wgt, e);
  k_embed<<<((n * HIDDEN) + B - 1) / B, B, 0, stream>>>(z, emb, X, n);

  int tiles   = (n + 15) / 16;
  int gblocks = (tiles + 7) / 8;   // 8 waves (16-row stripes) per 256-thread block
  int ablocks = (n + 7) / 8;       // 8 nodes (waves) per 256-thread block
  for (int l = 0; l < NUM_LAYERS; ++l) {
    k_gemm<<<gblocks, 256, 0, stream>>>(X, Ws + (size_t)l * HIDDEN * HIDDEN, H, n);
    k_aggregate<<<ablocks, 256, 0, stream>>>(H, rowptr, col, wgt,
                                             bs + (size_t)l * HIDDEN, X, n);
  }

  k_zero<<<((ng * HIDDEN) + B - 1) / B, B, 0, stream>>>(G, ng * HIDDEN);
  k_pool<<<((n * HIDDEN) + B - 1) / B, B, 0, stream>>>(X, batch, G, n);
  k_mlp<<<ng, 128, 0, stream>>>(G, W1, b1, W2, b2, out);
}